// DAGR_89429809037369
// MI455X (gfx1250) — compile-verified
//
#include <hip/hip_runtime.h>

#define CONF_THRES 0.05f
#define ROW_F 85          // floats per row: 4 box + 1 conf + 80 classes
#define NCLS 80
#define ROWS_PER_BLOCK 128
#define THREADS 128

#if defined(__AMDGCN__) && __has_builtin(__builtin_amdgcn_global_load_async_to_lds_b128) && \
    __has_builtin(__builtin_amdgcn_s_wait_asynccnt)
#define USE_ASYNC_LDS 1
#else
#define USE_ASYNC_LDS 0
#endif

// cpol: TH[2:0]=1 -> non-temporal load (input is 272MB > 192MB L2, touched once)
#define ASYNC_CPOL 1

typedef int v4i __attribute__((ext_vector_type(4)));
typedef __attribute__((address_space(1))) v4i* gv4_t;
typedef __attribute__((address_space(3))) v4i* lv4_t;

// Full-block staging constants: 128 rows * 85 floats = 10880 floats = 2720 x 16B
constexpr int kChunksFull = (ROWS_PER_BLOCK * ROW_F) / 4;           // 2720
constexpr int kFullIters  = kChunksFull / THREADS;                  // 21
constexpr int kTailChunks = kChunksFull - kFullIters * THREADS;     // 32

__global__ __launch_bounds__(THREADS) void DAGR_postproc_kernel(
    const float* __restrict__ pred, float* __restrict__ out, long long rows) {
  __shared__ float smem[ROWS_PER_BLOCK * ROW_F];  // 43,520 B -> ~7 blocks/WGP in 320KB LDS

  const long long rowBase = (long long)blockIdx.x * ROWS_PER_BLOCK;
  long long rem = rows - rowBase;
  if (rem <= 0) return;
  const int rowsHere = rem < ROWS_PER_BLOCK ? (int)rem : ROWS_PER_BLOCK;

  const float* gsrc = pred + rowBase * ROW_F;  // block base = 43,520B multiple -> 16B aligned
  const int tid = threadIdx.x;

#if USE_ASYNC_LDS
  if (rowsHere == ROWS_PER_BLOCK) {
    // Fully-unrolled staging: one base address, constant offsets folded into
    // the 24-bit IOFFSET of each global_load_async_to_lds_b128.
    gv4_t gp = (gv4_t)gsrc + tid;
    lv4_t lp = (lv4_t)smem + tid;
#pragma unroll
    for (int k = 0; k < kFullIters; ++k)
      __builtin_amdgcn_global_load_async_to_lds_b128(gp + k * THREADS,
                                                     lp + k * THREADS,
                                                     0, ASYNC_CPOL);
    if (tid < kTailChunks)
      __builtin_amdgcn_global_load_async_to_lds_b128(gp + kFullIters * THREADS,
                                                     lp + kFullIters * THREADS,
                                                     0, ASYNC_CPOL);
    __builtin_amdgcn_s_wait_asynccnt(0);
  } else {
    // Ragged final block (dead for 800,000 rows, kept for generality).
    const int totalFloats = rowsHere * ROW_F;
    const int nVec = totalFloats >> 2;
    for (int i = tid; i < nVec; i += THREADS)
      __builtin_amdgcn_global_load_async_to_lds_b128(
          (gv4_t)(gsrc + (size_t)i * 4), (lv4_t)(smem + i * 4), 0, ASYNC_CPOL);
    for (int i = (nVec << 2) + tid; i < totalFloats; i += THREADS)
      smem[i] = gsrc[i];
    __builtin_amdgcn_s_wait_asynccnt(0);
  }
#else
  {
    const int totalFloats = rowsHere * ROW_F;
    const int nVec = totalFloats >> 2;
    for (int i = tid; i < nVec; i += THREADS) {
      const float4 v = *(const float4*)(gsrc + (size_t)i * 4);
      *(float4*)(smem + i * 4) = v;
    }
    for (int i = (nVec << 2) + tid; i < totalFloats; i += THREADS)
      smem[i] = gsrc[i];
  }
#endif
  __syncthreads();

  if (tid < rowsHere) {
    // Row stride 85 words is coprime with 64 banks -> conflict-free LDS reads.
    const float* rp = smem + tid * ROW_F;
    const float x = rp[0], y = rp[1], w = rp[2], h = rp[3], conf = rp[4];

    float best = rp[5];
    int bi = 0;
#pragma unroll 8
    for (int j = 1; j < NCLS; ++j) {
      const float s = rp[5 + j];
      if (s > best) { best = s; bi = j; }  // strict > keeps first-max argmax semantics
    }

    const float validf = (conf >= CONF_THRES) ? 1.0f : 0.0f;
    const long long r = rowBase + tid;
    float* det = out + r * 7;
    __builtin_nontemporal_store(x * validf, det + 0);
    __builtin_nontemporal_store(y * validf, det + 1);
    __builtin_nontemporal_store(w * validf, det + 2);
    __builtin_nontemporal_store(h * validf, det + 3);
    __builtin_nontemporal_store(conf * validf, det + 4);
    __builtin_nontemporal_store(best * validf, det + 5);
    __builtin_nontemporal_store((float)bi * validf, det + 6);
    __builtin_nontemporal_store(validf, out + rows * 7 + r);  // valid mask
  }
}

extern "C" void kernel_launch(void* const* d_in, const int* in_sizes, int n_in,
                              void* d_out, int out_size, void* d_ws, size_t ws_size,
                              hipStream_t stream) {
  const float* pred = (const float*)d_in[0];
  float* out = (float*)d_out;
  const long long rows = (long long)in_sizes[0] / ROW_F;  // 8 * 100000
  const int blocks = (int)((rows + ROWS_PER_BLOCK - 1) / ROWS_PER_BLOCK);
  DAGR_postproc_kernel<<<blocks, THREADS, 0, stream>>>(pred, out, rows);
}